// RecStoreEmbeddingBagCollection_52922587021462
// MI455X (gfx1250) — compile-verified
//
#include <hip/hip_runtime.h>

// ---------------------------------------------------------------------------
// RecStore EmbeddingBagCollection: jagged gather + mean pool.
// B=2048, F=26, L=20, V=1e6, D=64. S=B*F segments, T=S*L ids.
// HBM-bound (~0.23 FLOP/byte): floor ~12.5us at 23.3 TB/s (less with L2 hits:
// ~655K distinct rows of 1.06M draws stay resident in the 192MB L2).
// CDNA5 path: GLOBAL_LOAD_ASYNC_TO_LDS_B128 gathers (ASYNCcnt-tracked,
// in-order completion -> split-wait software pipeline), wave32 mapping:
// 1 wave per bag, half-wave per 256B row, 2 dims per lane.
// ---------------------------------------------------------------------------

#define WAVES_PER_BLOCK 8
#define MAX_CHUNK 20   // rows staged in LDS per wave per pass
#define EMB_D 64       // embedding dim (floats)

// ---- Kernel 1: per-256-block exclusive scan of lengths ---------------------
__global__ void seg_scan_block(const int* __restrict__ lengths,
                               int* __restrict__ seg_off,
                               int* __restrict__ partials, int S) {
    __shared__ int sm[256];
    const int t = threadIdx.x;
    const int i = blockIdx.x * 256 + t;
    const int v = (i < S) ? lengths[i] : 0;
    sm[t] = v;
    __syncthreads();
    #pragma unroll
    for (int off = 1; off < 256; off <<= 1) {
        int x = (t >= off) ? sm[t - off] : 0;
        __syncthreads();
        sm[t] += x;
        __syncthreads();
    }
    const int incl = sm[t];
    if (i < S) seg_off[i] = incl - v;          // exclusive, block-local
    if (t == 255) partials[blockIdx.x] = incl; // block total
}

// ---- Kernel 2: scan block totals (single block, serial carry over chunks) --
__global__ void seg_scan_partials(int* __restrict__ partials, int nb) {
    __shared__ int sm[256];
    __shared__ int carry;
    const int t = threadIdx.x;
    if (t == 0) carry = 0;
    __syncthreads();
    for (int base = 0; base < nb; base += 256) {
        const int i = base + t;
        const int v = (i < nb) ? partials[i] : 0;
        sm[t] = v;
        __syncthreads();
        #pragma unroll
        for (int off = 1; off < 256; off <<= 1) {
            int x = (t >= off) ? sm[t - off] : 0;
            __syncthreads();
            sm[t] += x;
            __syncthreads();
        }
        const int incl = sm[t];
        if (i < nb) partials[i] = (incl - v) + carry;  // exclusive block base
        __syncthreads();
        if (t == 255) carry += incl;                    // chunk grand total
        __syncthreads();
    }
}

// ---- Kernel 3: gather + mean pool ------------------------------------------
__global__ __launch_bounds__(256) void embbag_pool(
    const int* __restrict__ values, const int* __restrict__ lengths,
    const float* __restrict__ table, const int* __restrict__ seg_off,
    const int* __restrict__ partials, float* __restrict__ out, int S) {

    // 8 waves x 20 rows x 256B staging = 40KB/WG (of 320KB per WGP)
    __shared__ __align__(16) float smem[WAVES_PER_BLOCK * MAX_CHUNK * EMB_D];

    const int lane = threadIdx.x & 31;
    const int wave = threadIdx.x >> 5;
    const int s = blockIdx.x * WAVES_PER_BLOCK + wave;
    if (s >= S) return;                       // wave-uniform exit

    const int len = lengths[s];
    const int off = seg_off[s] + partials[s >> 8];

    const int half = lane >> 4;               // which row of the pair
    const int sub  = lane & 15;               // 16B chunk within 256B row
    // Low 32 bits of a generic LDS pointer == wave-relative LDS byte offset
    // (flat aperture truncation, ISA 10.2), which is what async VDST wants.
    const unsigned ldsWave =
        (unsigned)(unsigned long long)&smem[wave * (MAX_CHUNK * EMB_D)];
    const unsigned long long tbase = (unsigned long long)table;
    // Lane owns dims {2*lane, 2*lane+1}: 32 lanes x 8B = all 64 LDS banks
    // exactly once -> conflict-free ds_load_b64 per row.
    const float* rowp = &smem[wave * (MAX_CHUNK * EMB_D) + lane * 2];

    float accx = 0.f, accy = 0.f;
    for (int base = 0; base < len; base += MAX_CHUNK) {
        const int cl = (len - base < MAX_CHUNK) ? (len - base) : MAX_CHUNK;

        // Stage indices: lane r holds row r's table index; broadcast via shfl.
        int myidx = (lane < cl) ? values[off + base + lane] : 0;

        // Issue up to 10 async B128 gathers: half-wave per row, 2 rows/op.
        // 16 lanes x 16B = one full 256B row -> perfectly coalesced.
        #pragma unroll
        for (int p = 0; p < MAX_CHUNK / 2; ++p) {
            const int row = 2 * p + half;
            if (row < cl) {
                const int ridx = __shfl(myidx, row, 32);
                const unsigned long long g =
                    tbase + (unsigned long long)(unsigned)ridx * 256ull
                          + (unsigned)(sub * 16);
                const unsigned l = ldsWave + (unsigned)(row * 256 + sub * 16);
                asm volatile("global_load_async_to_lds_b128 %0, %1, off"
                             :: "v"(l), "v"(g) : "memory");
            }
        }

        if (cl == MAX_CHUNK) {
            // Full chunk: exactly 10 ops in flight, completing in order.
            // Software pipeline: drain to <=5 (ops 0..4 = rows 0..9 landed),
            // pool them while ops 5..9 (rows 10..19) are still in flight.
            asm volatile("s_wait_asynccnt 0x5" ::: "memory");
            #pragma unroll
            for (int r = 0; r < MAX_CHUNK / 2; ++r) {
                const float2 v = *(const float2*)(rowp + r * EMB_D);
                accx += v.x;
                accy += v.y;
            }
            asm volatile("s_wait_asynccnt 0x0" ::: "memory");
            #pragma unroll
            for (int r = MAX_CHUNK / 2; r < MAX_CHUNK; ++r) {
                const float2 v = *(const float2*)(rowp + r * EMB_D);
                accx += v.x;
                accy += v.y;
            }
        } else {
            // Jagged tail: EXEC-predicated issue count is not architecturally
            // guaranteed, so drain fully before pooling.
            asm volatile("s_wait_asynccnt 0x0" ::: "memory");
            for (int r = 0; r < cl; ++r) {
                const float2 v = *(const float2*)(rowp + r * EMB_D);
                accx += v.x;
                accy += v.y;
            }
        }
    }

    const float inv = (len > 0) ? (1.0f / (float)len) : 0.0f;
    float2 res;
    res.x = accx * inv;
    res.y = accy * inv;
    *(float2*)&out[(long long)s * EMB_D + lane * 2] = res;  // coalesced b64
}

// ---------------------------------------------------------------------------
extern "C" void kernel_launch(void* const* d_in, const int* in_sizes, int n_in,
                              void* d_out, int out_size, void* d_ws,
                              size_t ws_size, hipStream_t stream) {
    const int*   values  = (const int*)d_in[0];   // [T] ids
    const int*   lengths = (const int*)d_in[1];   // [S]
    const float* table   = (const float*)d_in[2]; // [V, 64]
    float*       out     = (float*)d_out;         // [S, 64] == [B, F, 64]

    const int S  = in_sizes[1];
    const int NB = (S + 255) / 256;

    int* seg_off  = (int*)d_ws;                        // [S] block-local excl
    int* partials = seg_off + ((S + 255) & ~255);      // [NB] block bases

    seg_scan_block<<<NB, 256, 0, stream>>>(lengths, seg_off, partials, S);
    seg_scan_partials<<<1, 256, 0, stream>>>(partials, NB);

    const int blocks = (S + WAVES_PER_BLOCK - 1) / WAVES_PER_BLOCK;
    embbag_pool<<<blocks, 256, 0, stream>>>(values, lengths, table, seg_off,
                                            partials, out, S);
    (void)n_in; (void)out_size; (void)ws_size; (void)in_sizes;
}